// BeamTransformerDecoder_45732811768482
// MI455X (gfx1250) — compile-verified
//
#include <hip/hip_runtime.h>
#include <cstdint>
#include <cstddef>

#define BEAMK 4
#define MAXLEN 64
#define SOS 1
#define EOS 2
#define PADT 0
#define NEGV -1e9f
#define BB 16
#define SS 512
#define DDIM 512
#define VV 32000

#define LDA 68    // 64+4 pad: A-frag b64 LDS reads -> 64 distinct banks
#define LDB 144   // 128+16 pad: rows k and k+2 are 32 banks apart -> no 2-way conflict

typedef float v2f __attribute__((ext_vector_type(2)));
typedef float v8f __attribute__((ext_vector_type(8)));

// ---------------------------------------------------------------------------
// LDS-blocked batched fp32 WMMA GEMM: C[b] = A[b] (MxK) * B (KxN) (+bias).
// Block (256 thr, 8 waves) computes a 64x128 tile of C. Per 64-K chunk:
// A 64x64 (17KB) and B 64x128 (37KB) staged in LDS, then each wave runs
// 4 x V_WMMA_F32_16X16X4_F32 per K-step on one A fragment (16x64 of C per
// wave). Requires M%64==0, N%128==0, K%64==0 (true for all uses here).
// Tile mapping is wave-uniform => EXEC all-ones for every WMMA.
// ---------------------------------------------------------------------------
__global__ __launch_bounds__(256) void wmma_gemm_tiled(
    const float* __restrict__ A, const float* __restrict__ B,
    const float* __restrict__ bias, float* __restrict__ C,
    int M, int N, int K, long long strideA, long long strideC)
{
  __shared__ float As[64 * LDA];   // 17408 B
  __shared__ float Bs[64 * LDB];   // 36864 B
  A += (long long)blockIdx.y * strideA;
  C += (long long)blockIdx.y * strideC;

  const int nblocks = N >> 7;                 // N / 128
  const int nb = blockIdx.x % nblocks;
  const int mb = blockIdx.x / nblocks;
  const int m_base = mb * 64;
  const int n_base = nb * 128;

  const int tid  = threadIdx.x;
  const int wave = tid >> 5;
  const int lane = tid & 31;
  const int hi = lane >> 4;                   // 0/1
  const int lo = lane & 15;                   // 0..15
  const int mt = wave & 3;                    // which 16-row subtile of the 64
  const int ng = wave >> 2;                   // which 64-col group of the 128

  v8f acc0 = {0.f,0.f,0.f,0.f,0.f,0.f,0.f,0.f};
  v8f acc1 = acc0, acc2 = acc0, acc3 = acc0;

  for (int kb = 0; kb < K; kb += 64) {
    // ---- stage A chunk: 64 rows x 64 k, float4 coalesced ----
    #pragma unroll
    for (int j = 0; j < 4; j++) {
      const int fid = tid + j * 256;          // 0..1023 float4 slots
      const int r  = fid >> 4;
      const int c4 = (fid & 15) << 2;
      const float4 v = *(const float4*)(A + (size_t)(m_base + r) * K + kb + c4);
      *(float4*)(&As[r * LDA + c4]) = v;      // LDA*4 % 16 == 0 -> aligned
    }
    // ---- stage B chunk: 64 k x 128 n, float4 coalesced ----
    #pragma unroll
    for (int j = 0; j < 8; j++) {
      const int fid = tid + j * 256;          // 0..2047 float4 slots
      const int r  = fid >> 5;
      const int c4 = (fid & 31) << 2;
      const float4 v = *(const float4*)(B + (size_t)(kb + r) * N + n_base + c4);
      *(float4*)(&Bs[r * LDB + c4]) = v;      // LDB*4 % 16 == 0 -> aligned
    }
    // prefetch next B chunk into cache while we compute this one
    if (kb + 64 < K) {
      const int r = tid >> 2;                 // 0..63
      const int c = (tid & 3) << 5;           // 0,32,64,96
      __builtin_prefetch(B + (size_t)(kb + 64 + r) * N + n_base + c, 0, 1);
    }
    __syncthreads();

    // ---- compute: 16 K-steps x 4 WMMA ----
    #pragma unroll 4
    for (int k = 0; k < 64; k += 4) {
      const float2 a2 = *(const float2*)(&As[(mt * 16 + lo) * LDA + k + 2 * hi]);
      v2f a; a.x = a2.x; a.y = a2.y;
      const int bbase = (k + 2 * hi) * LDB + ng * 64 + lo;
      v2f b0, b1, b2, b3;
      b0.x = Bs[bbase];      b0.y = Bs[bbase + LDB];
      b1.x = Bs[bbase + 16]; b1.y = Bs[bbase + LDB + 16];
      b2.x = Bs[bbase + 32]; b2.y = Bs[bbase + LDB + 32];
      b3.x = Bs[bbase + 48]; b3.y = Bs[bbase + LDB + 48];
      acc0 = __builtin_amdgcn_wmma_f32_16x16x4_f32(false, a, false, b0, (short)0, acc0, false, false);
      acc1 = __builtin_amdgcn_wmma_f32_16x16x4_f32(false, a, false, b1, (short)0, acc1, false, false);
      acc2 = __builtin_amdgcn_wmma_f32_16x16x4_f32(false, a, false, b2, (short)0, acc2, false, false);
      acc3 = __builtin_amdgcn_wmma_f32_16x16x4_f32(false, a, false, b3, (short)0, acc3, false, false);
    }
    __syncthreads();
  }

  // ---- epilogue: C layout vgpr i -> M=i (lo lanes) / M=i+8 (hi lanes) ----
  const int colb = n_base + ng * 64 + lo;
  const float bv0 = bias ? bias[colb]      : 0.f;
  const float bv1 = bias ? bias[colb + 16] : 0.f;
  const float bv2 = bias ? bias[colb + 32] : 0.f;
  const float bv3 = bias ? bias[colb + 48] : 0.f;
  #pragma unroll
  for (int i = 0; i < 8; i++) {
    const int row = m_base + mt * 16 + i + 8 * hi;
    float* cr = C + (size_t)row * N + colb;
    cr[0]  = acc0[i] + bv0;
    cr[16] = acc1[i] + bv1;
    cr[32] = acc2[i] + bv2;
    cr[48] = acc3[i] + bv3;
  }
}

// ---------------------------------------------------------------------------
// Per-batch attention step: e = emb[tok]; q = e@Wq; att = q@encK^T (masked);
// softmax over S; ctx = att@enc; X = e + ctx.  One block per batch b,
// 4 beam-groups of 64 lanes for the softmax part.
// ---------------------------------------------------------------------------
__global__ __launch_bounds__(256) void att_ctx_kernel(
    const float* __restrict__ emb, const float* __restrict__ Wq,
    const float* __restrict__ encK, const float* __restrict__ enc,
    const int* __restrict__ enc_len, const int* __restrict__ last_tok,
    float* __restrict__ X)
{
  __shared__ float e_s[BEAMK * DDIM];
  __shared__ float q_s[BEAMK * DDIM];
  __shared__ float p_s[BEAMK * SS];
  __shared__ float red[256];
  __shared__ float ssum[BEAMK];

  const int b = blockIdx.x;
  const int tid = threadIdx.x;

  // 1) embedding gather into LDS
  for (int idx = tid; idx < BEAMK * DDIM; idx += 256) {
    const int k = idx >> 9, d = idx & (DDIM - 1);
    const int tok = last_tok[b * BEAMK + k];
    e_s[idx] = emb[(size_t)tok * DDIM + d];
  }
  __syncthreads();

  // 2) q = e @ Wq
  for (int idx = tid; idx < BEAMK * DDIM; idx += 256) {
    const int k = idx >> 9, d = idx & (DDIM - 1);
    float acc = 0.f;
    for (int dd = 0; dd < DDIM; dd++)
      acc += e_s[k * DDIM + dd] * Wq[(size_t)dd * DDIM + d];
    q_s[idx] = acc;
  }
  __syncthreads();

  // 3) masked attention scores + softmax (group g = beam, 64 lanes each)
  const float scale = 0.044194173824159216f;  // 1/sqrt(512)
  const int g = tid >> 6, l = tid & 63;
  const int len = enc_len[b];
  float av[8];
  float lmax = -3.4e38f;
  #pragma unroll
  for (int i = 0; i < 8; i++) {
    const int s = l + i * 64;
    const float* kr = encK + ((size_t)b * SS + s) * DDIM;
    float acc = 0.f;
    for (int dd = 0; dd < DDIM; dd++) acc += q_s[g * DDIM + dd] * kr[dd];
    acc *= scale;
    if (s >= len) acc = NEGV;
    av[i] = acc;
    lmax = fmaxf(lmax, acc);
  }
  red[tid] = lmax; __syncthreads();
  for (int off = 32; off >= 1; off >>= 1) {
    if (l < off) red[tid] = fmaxf(red[tid], red[tid + off]);
    __syncthreads();
  }
  const float gmax = red[g * 64];
  __syncthreads();
  float lsum = 0.f;
  #pragma unroll
  for (int i = 0; i < 8; i++) {
    const float e = __expf(av[i] - gmax);
    p_s[g * SS + l + i * 64] = e;
    lsum += e;
  }
  red[tid] = lsum; __syncthreads();
  for (int off = 32; off >= 1; off >>= 1) {
    if (l < off) red[tid] += red[tid + off];
    __syncthreads();
  }
  if (l == 0) ssum[g] = red[tid];
  __syncthreads();

  // 4) ctx = p @ enc, X = e + ctx/sum
  for (int idx = tid; idx < BEAMK * DDIM; idx += 256) {
    const int k = idx >> 9, d = idx & (DDIM - 1);
    const float inv = 1.f / ssum[k];
    float acc = 0.f;
    for (int s = 0; s < SS; s++)
      acc += p_s[k * SS + s] * enc[((size_t)b * SS + s) * DDIM + d];
    X[(size_t)(b * BEAMK + k) * DDIM + d] = e_s[k * DDIM + d] + acc * inv;
  }
}

// ---------------------------------------------------------------------------
// Per-row (64 rows): max, log-sum-exp, and top-4 logits -> candidate
// log-probs + tokens.  logp = logit - max - log(sum exp(logit - max)).
// ---------------------------------------------------------------------------
__global__ __launch_bounds__(256) void reduce_top4_kernel(
    const float* __restrict__ logits, float* __restrict__ cand_lp,
    int* __restrict__ cand_tok)
{
  __shared__ float red[256];
  __shared__ float sval[1024];
  __shared__ int   sidx[1024];
  const int row = blockIdx.x;
  const int tid = threadIdx.x;
  const float* rp = logits + (size_t)row * VV;

  float tv[4] = {-3.4e38f, -3.4e38f, -3.4e38f, -3.4e38f};
  int   ti[4] = {0, 0, 0, 0};
  float lmax = -3.4e38f;
  for (int n = tid; n < VV; n += 256) {
    const float x = rp[n];
    lmax = fmaxf(lmax, x);
    if (x > tv[3]) {
      tv[3] = x; ti[3] = n;
      #pragma unroll
      for (int j = 3; j > 0; j--) {
        if (tv[j] > tv[j - 1]) {
          float fv = tv[j]; tv[j] = tv[j - 1]; tv[j - 1] = fv;
          int   iv = ti[j]; ti[j] = ti[j - 1]; ti[j - 1] = iv;
        }
      }
    }
  }
  red[tid] = lmax; __syncthreads();
  for (int off = 128; off >= 1; off >>= 1) {
    if (tid < off) red[tid] = fmaxf(red[tid], red[tid + off]);
    __syncthreads();
  }
  const float gmax = red[0];
  __syncthreads();
  float lsum = 0.f;
  for (int n = tid; n < VV; n += 256) lsum += __expf(rp[n] - gmax);
  red[tid] = lsum; __syncthreads();
  for (int off = 128; off >= 1; off >>= 1) {
    if (tid < off) red[tid] += red[tid + off];
    __syncthreads();
  }
  const float gsum = red[0];
  #pragma unroll
  for (int j = 0; j < 4; j++) { sval[tid * 4 + j] = tv[j]; sidx[tid * 4 + j] = ti[j]; }
  __syncthreads();
  if (tid == 0) {
    const float corr = gmax + __logf(gsum);
    for (int slot = 0; slot < 4; slot++) {
      int bi = 0; float bv = sval[0];
      for (int i = 1; i < 1024; i++)
        if (sval[i] > bv) { bv = sval[i]; bi = i; }
      cand_lp[row * 4 + slot] = bv - corr;
      cand_tok[row * 4 + slot] = sidx[bi];
      sval[bi] = -3.4e38f;
    }
  }
}

// ---------------------------------------------------------------------------
// Beam combine/update. One thread per batch. is_init: first expansion of SOS.
// Ping-pong sequence buffers handle the beam-gather permutation.
// ---------------------------------------------------------------------------
__global__ void beam_step_kernel(
    float* __restrict__ scores, int* __restrict__ finished,
    int* __restrict__ last_tok, const float* __restrict__ cand_lp,
    const int* __restrict__ cand_tok, const int* __restrict__ seqs_src,
    int* __restrict__ seqs_dst, int t, int is_init)
{
  const int b = threadIdx.x;
  if (b >= BB) return;
  if (is_init) {
    const int row = b * BEAMK;  // all beams identical (SOS); use beam 0's row
    for (int k = 0; k < BEAMK; k++) {
      const int tok = cand_tok[row * 4 + k];
      scores[b * BEAMK + k] = cand_lp[row * 4 + k];
      finished[b * BEAMK + k] = (tok == EOS) ? 1 : 0;
      last_tok[b * BEAMK + k] = tok;
      int* sd = seqs_dst + (b * BEAMK + k) * MAXLEN;
      for (int tt = 0; tt < MAXLEN; tt++) sd[tt] = PADT;
      sd[0] = SOS; sd[1] = tok;
    }
    return;
  }
  float sc[BEAMK]; int fin[BEAMK];
  for (int k = 0; k < BEAMK; k++) {
    sc[k] = scores[b * BEAMK + k];
    fin[k] = finished[b * BEAMK + k];
  }
  float tot[16]; int vv[16];
  for (int k = 0; k < BEAMK; k++)
    for (int j = 0; j < BEAMK; j++) {
      float lp; int v;
      if (fin[k]) { lp = (j == 0) ? 0.f : NEGV; v = PADT; }
      else {
        lp = cand_lp[(b * BEAMK + k) * 4 + j];
        v  = cand_tok[(b * BEAMK + k) * 4 + j];
      }
      tot[k * 4 + j] = sc[k] + lp;
      vv[k * 4 + j] = v;
    }
  float nsc[BEAMK]; int nbi[BEAMK], ntok[BEAMK];
  for (int slot = 0; slot < BEAMK; slot++) {
    int bi = 0; float bv = tot[0];
    for (int i = 1; i < 16; i++) if (tot[i] > bv) { bv = tot[i]; bi = i; }
    nsc[slot] = bv; nbi[slot] = bi >> 2; ntok[slot] = vv[bi];
    tot[bi] = -3.4e38f;
  }
  for (int slot = 0; slot < BEAMK; slot++) {
    const int* ss = seqs_src + (b * BEAMK + nbi[slot]) * MAXLEN;
    int* sd = seqs_dst + (b * BEAMK + slot) * MAXLEN;
    for (int tt = 0; tt < MAXLEN; tt++) sd[tt] = (tt == t) ? ntok[slot] : ss[tt];
    scores[b * BEAMK + slot] = nsc[slot];
    finished[b * BEAMK + slot] = (fin[nbi[slot]] || ntok[slot] == EOS) ? 1 : 0;
    last_tok[b * BEAMK + slot] = ntok[slot];
  }
}

__global__ void init_state_kernel(int* __restrict__ last_tok)
{
  const int i = threadIdx.x;
  if (i < BB * BEAMK) last_tok[i] = SOS;
}

__global__ void finalize_kernel(const float* __restrict__ scores,
                                const int* __restrict__ seqs,
                                float* __restrict__ out)
{
  const int b = threadIdx.x;
  if (b >= BB) return;
  int best = 0; float bv = scores[b * BEAMK];
  for (int k = 1; k < BEAMK; k++)
    if (scores[b * BEAMK + k] > bv) { bv = scores[b * BEAMK + k]; best = k; }
  const int* s = seqs + (b * BEAMK + best) * MAXLEN;
  for (int tt = 0; tt < MAXLEN; tt++) out[b * MAXLEN + tt] = (float)s[tt];
  out[BB * MAXLEN + b] = bv;
}

// ---------------------------------------------------------------------------
extern "C" void kernel_launch(void* const* d_in, const int* in_sizes, int n_in,
                              void* d_out, int out_size, void* d_ws, size_t ws_size,
                              hipStream_t stream) {
  (void)in_sizes; (void)n_in; (void)out_size; (void)ws_size;
  const float* enc     = (const float*)d_in[0];   // [16,512,512]
  const int*   enc_len = (const int*)  d_in[1];   // [16]
  const float* emb     = (const float*)d_in[2];   // [32000,512]
  const float* Wq      = (const float*)d_in[3];   // [512,512]
  const float* Wk      = (const float*)d_in[4];   // [512,512]
  const float* Wfc     = (const float*)d_in[5];   // [512,32000]
  const float* bfc     = (const float*)d_in[6];   // [32000]
  float* out = (float*)d_out;                     // 16*64 tokens (as float) + 16 scores

  // Workspace carve-up (256B aligned)
  char* ws = (char*)d_ws;
  size_t off = 0;
  auto carve = [&](size_t bytes) -> char* {
    char* p = ws + off;
    off = (off + bytes + 255) & ~(size_t)255;
    return p;
  };
  float* encK     = (float*)carve((size_t)BB * SS * DDIM * 4);      // 16.78 MB
  float* X        = (float*)carve((size_t)BB * BEAMK * DDIM * 4);   // 128 KB
  float* logits   = (float*)carve((size_t)BB * BEAMK * VV * 4);     // 8.19 MB
  float* cand_lp  = (float*)carve((size_t)BB * BEAMK * 4 * 4);
  int*   cand_tok = (int*)  carve((size_t)BB * BEAMK * 4 * 4);
  float* scores   = (float*)carve((size_t)BB * BEAMK * 4);
  int*   finished = (int*)  carve((size_t)BB * BEAMK * 4);
  int*   last_tok = (int*)  carve((size_t)BB * BEAMK * 4);
  int*   seqs0    = (int*)  carve((size_t)BB * BEAMK * MAXLEN * 4);
  int*   seqs1    = (int*)  carve((size_t)BB * BEAMK * MAXLEN * 4);

  // init: all beams start at SOS
  init_state_kernel<<<1, 64, 0, stream>>>(last_tok);

  // encK[b] = enc[b] @ Wk : M=512, N=512, K=512, 16 batches.
  // block tile 64x128 -> grid.x = (512/128)*(512/64) = 32
  wmma_gemm_tiled<<<dim3((DDIM / 128) * (SS / 64), BB), 256, 0, stream>>>(
      enc, Wk, nullptr, encK, SS, DDIM, DDIM,
      (long long)SS * DDIM, (long long)SS * DDIM);

  for (int t = 1; t < MAXLEN; t++) {
    // x = e + ctx   -> X [64,512]
    att_ctx_kernel<<<BB, 256, 0, stream>>>(emb, Wq, encK, enc, enc_len,
                                           last_tok, X);
    // logits = X @ Wfc + bfc : M=64, N=32000, K=512
    // block tile 64x128 -> grid.x = (32000/128)*(64/64) = 250
    wmma_gemm_tiled<<<dim3((VV / 128) * ((BB * BEAMK) / 64), 1), 256, 0, stream>>>(
        X, Wfc, bfc, logits, BB * BEAMK, VV, DDIM, 0LL, 0LL);
    // per-row log-softmax stats + top-4
    reduce_top4_kernel<<<BB * BEAMK, 256, 0, stream>>>(logits, cand_lp, cand_tok);
    // beam combine + gather (ping-pong sequence buffers)
    int* src = (t & 1) ? seqs0 : seqs1;
    int* dst = (t & 1) ? seqs1 : seqs0;
    beam_step_kernel<<<1, BB, 0, stream>>>(scores, finished, last_tok,
                                           cand_lp, cand_tok, src, dst,
                                           t, (t == 1) ? 1 : 0);
  }
  // t ran to 63 (odd) -> final sequences live in seqs1
  finalize_kernel<<<1, BB, 0, stream>>>(scores, seqs1, out);
}